// MHLV_86414741996210
// MI455X (gfx1250) — compile-verified
//
#include <hip/hip_runtime.h>
#include <math.h>

// Problem constants (from the reference): B=8 scenes, A=16 agents, L=80 lanes,
// D=128, H=6 heads. N = 768 nodes, HD = H*D = 768. Per scene: 96 nodes, dense.
#define NNODES 768
#define NAGT   128
#define DIM    128
#define HD     768
#define NSC    96   // nodes per scene

typedef __attribute__((ext_vector_type(2))) float v2f;
typedef __attribute__((ext_vector_type(8))) float v8f;

// Map scene-local index (0..95) -> global node row.
__device__ __forceinline__ int node_of(int b, int i) {
  return (i < 16) ? (b * 16 + i) : (128 + b * 80 + (i - 16));
}

// One wave computes a 16x16 f32 tile of C += A(16xK, row-major lda) * B(Kx16, row-major ldb)
// using V_WMMA_F32_16X16X4_F32.
// A frag (16x4): lane&15 = M row; lanes 0-15 hold K={k,k+1}, lanes 16-31 hold K={k+2,k+3}.
// B frag (4x16): lane&15 = N col; same K split across lane halves.
__device__ __forceinline__ v8f gemm_tile_rr(const float* __restrict__ A, int lda,
                                            const float* __restrict__ B, int ldb,
                                            int Kdim, v8f acc) {
  const int lane = threadIdx.x & 31;
  const int half = lane >> 4;
  const int idx  = lane & 15;
  const float* arow = A + idx * lda + 2 * half;
  const float* bcol = B + idx + (2 * half) * ldb;
  for (int k0 = 0; k0 < Kdim; k0 += 4) {
    v2f a, b;
    a.x = arow[k0];
    a.y = arow[k0 + 1];
    b.x = bcol[k0 * ldb];
    b.y = bcol[k0 * ldb + ldb];
    acc = __builtin_amdgcn_wmma_f32_16x16x4_f32(false, a, false, b, (short)0, acc,
                                                false, false);
  }
  return acc;
}

// Store one 16x16 f32 C/D tile (VGPR r: lanes0-15 -> M=r, lanes16-31 -> M=r+8).
__device__ __forceinline__ void store_tile(float* __restrict__ C, int ldc, v8f acc,
                                           bool do_relu) {
  const int lane = threadIdx.x & 31;
  const int half = lane >> 4;
  const int n    = lane & 15;
#pragma unroll
  for (int r = 0; r < 8; ++r) {
    const int m = r + 8 * half;
    float v = acc[r];
    if (do_relu) v = fmaxf(v, 0.0f);
    C[m * ldc + n] = v;
  }
}

// ---------------------------------------------------------------------------
// K0: nodes = concat(agents, lanes)   [768 x 128]
__global__ void k_concat(const float* __restrict__ agents,
                         const float* __restrict__ lanes,
                         float* __restrict__ nodes) {
  const int i = blockIdx.x * blockDim.x + threadIdx.x;
  const int NA = NAGT * DIM;              // 16384
  const int NT = NNODES * DIM;            // 98304
  if (i < NA)      nodes[i] = agents[i];
  else if (i < NT) nodes[i] = lanes[i - NA];
}

// ---------------------------------------------------------------------------
// K1: Q/K/V = nodes @ W{q,k,v}  (relu on V).  M=768, N=768, K=128.
// blockIdx.y selects which projection; 8 waves/block, 1 tile/wave.
__global__ void __launch_bounds__(256) k_qkv(const float* __restrict__ nodes,
                                             const float* __restrict__ Wq,
                                             const float* __restrict__ Wk,
                                             const float* __restrict__ Wv,
                                             float* __restrict__ Qo,
                                             float* __restrict__ Ko,
                                             float* __restrict__ Vo) {
  const int which = blockIdx.y;
  const float* W = (which == 0) ? Wq : (which == 1) ? Wk : Wv;
  float*       O = (which == 0) ? Qo : (which == 1) ? Ko : Vo;
  const int wave = threadIdx.x >> 5;
  const int tile = blockIdx.x * 8 + wave;        // 0..2303
  const int tm = tile / 48, tn = tile % 48;
  v8f acc = {};
  acc = gemm_tile_rr(nodes + tm * 16 * DIM, DIM, W + tn * 16, HD, DIM, acc);
  store_tile(O + (tm * 16) * HD + tn * 16, HD, acc, which == 2);
}

// ---------------------------------------------------------------------------
// K2: per (scene b, head h) dense attention.
// S = Qs Ks^T * scale (96x96, K=128) -> row softmax in LDS -> O = S Vs (96x128, K=96).
__global__ void __launch_bounds__(256) k_attn(const float* __restrict__ Q,
                                              const float* __restrict__ Km,
                                              const float* __restrict__ V,
                                              float* __restrict__ Out) {
  __shared__ float S[NSC][NSC + 4];       // 96x100 f32 = 38.4 KB
  const int b    = blockIdx.x;            // scene
  const int h    = blockIdx.y;            // head
  const int wave = threadIdx.x >> 5;
  const int lane = threadIdx.x & 31;
  const int half = lane >> 4;
  const int idx  = lane & 15;
  const float scale = 0.088388347648318447f;   // 128^-0.5

  // --- scores: 6x6 tiles of 16x16, 8 waves round-robin ---
  for (int t = wave; t < 36; t += 8) {
    const int tm = t / 6, tn = t % 6;
    const float* arow = Q  + (size_t)node_of(b, tm * 16 + idx) * HD + h * DIM + 2 * half;
    const float* brow = Km + (size_t)node_of(b, tn * 16 + idx) * HD + h * DIM + 2 * half;
    v8f acc = {};
    for (int k0 = 0; k0 < DIM; k0 += 4) {
      v2f a, bb;
      a.x  = arow[k0];  a.y  = arow[k0 + 1];
      bb.x = brow[k0];  bb.y = brow[k0 + 1];   // B[k][n] = K[node(n)][h*128+k]
      acc = __builtin_amdgcn_wmma_f32_16x16x4_f32(false, a, false, bb, (short)0, acc,
                                                  false, false);
    }
#pragma unroll
    for (int r = 0; r < 8; ++r)
      S[tm * 16 + r + 8 * half][tn * 16 + idx] = acc[r] * scale;
  }
  __syncthreads();

  // --- row softmax (global-max shift in reference cancels; per-row is identical) ---
  if (threadIdx.x < NSC) {
    float* row = S[threadIdx.x];
    float mx = -3.4e38f;
    for (int j = 0; j < NSC; ++j) mx = fmaxf(mx, row[j]);
    float sum = 0.0f;
    for (int j = 0; j < NSC; ++j) { float e = __expf(row[j] - mx); row[j] = e; sum += e; }
    const float inv = 1.0f / sum;
    for (int j = 0; j < NSC; ++j) row[j] *= inv;
  }
  __syncthreads();

  // --- O = P * V : 6x8 tiles, A from LDS, B gathered from V ---
  for (int t = wave; t < 48; t += 8) {
    const int tm = t >> 3, tn = t & 7;
    v8f acc = {};
    for (int k0 = 0; k0 < NSC; k0 += 4) {
      const int kk = k0 + 2 * half;
      v2f a, bb;
      a.x  = S[tm * 16 + idx][kk];
      a.y  = S[tm * 16 + idx][kk + 1];
      bb.x = V[(size_t)node_of(b, kk)     * HD + h * DIM + tn * 16 + idx];
      bb.y = V[(size_t)node_of(b, kk + 1) * HD + h * DIM + tn * 16 + idx];
      acc = __builtin_amdgcn_wmma_f32_16x16x4_f32(false, a, false, bb, (short)0, acc,
                                                  false, false);
    }
#pragma unroll
    for (int r = 0; r < 8; ++r)
      Out[(size_t)node_of(b, tm * 16 + r + 8 * half) * HD + h * DIM + tn * 16 + idx] = acc[r];
  }
}

// ---------------------------------------------------------------------------
// K3: T1 = relu(AttOut @ Wout1)   M=768, N=128, K=768.  48 blocks x 8 waves.
__global__ void __launch_bounds__(256) k_out1(const float* __restrict__ AO,
                                              const float* __restrict__ Wout1,
                                              float* __restrict__ T1) {
  const int tm = blockIdx.x;              // 0..47
  const int tn = threadIdx.x >> 5;        // 0..7
  v8f acc = {};
  acc = gemm_tile_rr(AO + tm * 16 * HD, HD, Wout1 + tn * 16, DIM, HD, acc);
  store_tile(T1 + tm * 16 * DIM + tn * 16, DIM, acc, true);
}

// ---------------------------------------------------------------------------
// K4: X = T1@Wout2 + nodes@W1; Y = relu(LayerNorm(X)*g + b). Fused per 16-row block.
__global__ void __launch_bounds__(256) k_ln(const float* __restrict__ T1,
                                            const float* __restrict__ Wout2,
                                            const float* __restrict__ nodes,
                                            const float* __restrict__ W1,
                                            const float* __restrict__ g,
                                            const float* __restrict__ bvec,
                                            float* __restrict__ Y) {
  __shared__ float X[16][DIM + 4];
  __shared__ float s_mu[16], s_rs[16];
  const int tm   = blockIdx.x;            // 0..47 (16-row block)
  const int tn   = threadIdx.x >> 5;      // column tile 0..7
  const int lane = threadIdx.x & 31;
  const int half = lane >> 4;
  const int idx  = lane & 15;

  v8f acc = {};
  acc = gemm_tile_rr(T1 + tm * 16 * DIM, DIM, Wout2 + tn * 16, DIM, DIM, acc);
  acc = gemm_tile_rr(nodes + tm * 16 * DIM, DIM, W1 + tn * 16, DIM, DIM, acc);
#pragma unroll
  for (int r = 0; r < 8; ++r)
    X[r + 8 * half][tn * 16 + idx] = acc[r];
  __syncthreads();

  if (threadIdx.x < 16) {
    const int r = threadIdx.x;
    float m = 0.0f;
    for (int c = 0; c < DIM; ++c) m += X[r][c];
    m *= (1.0f / DIM);
    float v = 0.0f;
    for (int c = 0; c < DIM; ++c) { float d = X[r][c] - m; v += d * d; }
    v *= (1.0f / DIM);
    s_mu[r] = m;
    s_rs[r] = rsqrtf(v + 1e-5f);
  }
  __syncthreads();

  for (int e = threadIdx.x; e < 16 * DIM; e += 256) {
    const int r = e >> 7, c = e & (DIM - 1);
    const float val = (X[r][c] - s_mu[r]) * s_rs[r] * g[c] + bvec[c];
    Y[(tm * 16 + r) * DIM + c] = fmaxf(val, 0.0f);
  }
}

// ---------------------------------------------------------------------------
// K5: out = relu(Y[:128] @ W2 + nodes[:128])  -> d_out [128 x 128]
__global__ void __launch_bounds__(256) k_final(const float* __restrict__ Y,
                                               const float* __restrict__ W2,
                                               const float* __restrict__ nodes,
                                               float* __restrict__ out) {
  const int tm   = blockIdx.x;            // 0..7
  const int tn   = threadIdx.x >> 5;      // 0..7
  const int lane = threadIdx.x & 31;
  const int half = lane >> 4;
  const int idx  = lane & 15;
  v8f acc = {};
  acc = gemm_tile_rr(Y + tm * 16 * DIM, DIM, W2 + tn * 16, DIM, DIM, acc);
#pragma unroll
  for (int r = 0; r < 8; ++r) {
    const int m = r + 8 * half;
    const float v = acc[r] + nodes[(tm * 16 + m) * DIM + tn * 16 + idx];
    out[(tm * 16 + m) * DIM + tn * 16 + idx] = fmaxf(v, 0.0f);
  }
}

// ---------------------------------------------------------------------------
extern "C" void kernel_launch(void* const* d_in, const int* in_sizes, int n_in,
                              void* d_out, int out_size, void* d_ws, size_t ws_size,
                              hipStream_t stream) {
  const float* agents = (const float*)d_in[0];
  const float* lanes  = (const float*)d_in[1];
  const float* Wq     = (const float*)d_in[2];
  const float* Wk     = (const float*)d_in[3];
  const float* Wv     = (const float*)d_in[4];
  const float* Wout1  = (const float*)d_in[5];
  const float* Wout2  = (const float*)d_in[6];
  const float* W1     = (const float*)d_in[7];
  const float* ln_g   = (const float*)d_in[8];
  const float* ln_b   = (const float*)d_in[9];
  const float* W2     = (const float*)d_in[10];
  // d_in[11]=hi, d_in[12]=wi: edge lists are the full per-scene cross product;
  // we exploit that block-dense structure directly and do not need them.
  (void)in_sizes; (void)n_in; (void)out_size; (void)ws_size;

  float* ws    = (float*)d_ws;
  float* nodes = ws;                                  //  98304
  float* Q     = nodes + NNODES * DIM;                // 589824
  float* Km    = Q     + NNODES * HD;                 // 589824
  float* V     = Km    + NNODES * HD;                 // 589824
  float* AO    = V     + NNODES * HD;                 // 589824
  float* T1    = AO    + NNODES * HD;                 //  98304
  float* Y     = T1    + NNODES * DIM;                //  98304
  float* outp  = (float*)d_out;

  k_concat<<<dim3((NNODES * DIM) / 256), 256, 0, stream>>>(agents, lanes, nodes);
  k_qkv   <<<dim3(288, 3), 256, 0, stream>>>(nodes, Wq, Wk, Wv, Q, Km, V);
  k_attn  <<<dim3(8, 6),   256, 0, stream>>>(Q, Km, V, AO);
  k_out1  <<<dim3(48),     256, 0, stream>>>(AO, Wout1, T1);
  k_ln    <<<dim3(48),     256, 0, stream>>>(T1, Wout2, nodes, W1, ln_g, ln_b, Y);
  k_final <<<dim3(8),      256, 0, stream>>>(Y, W2, nodes, outp);
}